// MMDLoss_79465484910887
// MI455X (gfx1250) — compile-verified
//
#include <hip/hip_runtime.h>
#include <math.h>

typedef __attribute__((ext_vector_type(2))) float v2f;
typedef __attribute__((ext_vector_type(4))) float v4f;
typedef __attribute__((ext_vector_type(8))) float v8f;
typedef __attribute__((ext_vector_type(4))) int   v4i;

#define NB 4          // batches
#define MHALF 1024    // M == N
#define PP 2048       // P = M+N
#define DD 512        // feature dim
#define TILES 16      // P / 128
#define TRI_TILES 136 // 16*17/2 upper-triangular 128x128 tiles
#define KT 16         // K-chunk staged in LDS
#define NC (DD / KT)  // 32 chunks
#define LSTR 20       // LDS row stride (16 + 4 pad) -> conflict-free ds_load_b64

#if __has_builtin(__builtin_amdgcn_global_load_async_to_lds_b128) && \
    __has_builtin(__builtin_amdgcn_s_wait_asynccnt)
#define USE_ASYNC 1
#else
#define USE_ASYNC 0
#endif

__device__ __forceinline__ const float* zrow(const float* __restrict__ in,
                                             const float* __restrict__ tar,
                                             int b, int row)
{
    return (row < MHALF) ? (in  + ((size_t)b * MHALF + row) * DD)
                         : (tar + ((size_t)b * MHALF + (row - MHALF)) * DD);
}

// 16B global -> LDS copy: async (ASYNCcnt / no VGPR round-trip) when available.
__device__ __forceinline__ void cp_b128(const float* g, float* l)
{
#if USE_ASYNC
    __builtin_amdgcn_global_load_async_to_lds_b128((v4i*)g, (v4i*)l, 0, 0);
#else
    *(v4f*)l = *(const v4f*)g;
#endif
}

__device__ __forceinline__ void cp_commit_barrier()
{
#if USE_ASYNC
    __builtin_amdgcn_s_wait_asynccnt(0);
#endif
    __syncthreads();
}

// ---------------------------------------------------------------------------
// Kernel 0: per-point z2[b,p] = -0.5*||z_p||^2 and signed weights w_cat.
// ---------------------------------------------------------------------------
__global__ void mmd_prep(const float* __restrict__ input,
                         const float* __restrict__ target,
                         const float* __restrict__ w_in,
                         const float* __restrict__ w_tar,
                         float* __restrict__ z2,
                         float* __restrict__ wcat)
{
    int gwave = (blockIdx.x * blockDim.x + threadIdx.x) >> 5;
    int lane  = threadIdx.x & 31;
    int b = gwave >> 11;
    int p = gwave & (PP - 1);

    const float* row = zrow(input, target, b, p);
    float s = 0.f;
#pragma unroll
    for (int i = 0; i < DD / 32; ++i) {
        float v = row[lane + 32 * i];
        s += v * v;
    }
#pragma unroll
    for (int off = 16; off > 0; off >>= 1)
        s += __shfl_xor(s, off, 32);

    if (lane == 0) {
        z2[gwave]   = -0.5f * s;
        wcat[gwave] = (p < MHALF) ? w_in[b * MHALF + p]
                                  : -w_tar[b * MHALF + (p - MHALF)];
    }
}

// ---------------------------------------------------------------------------
// Kernel 1: fused Gram-tile (f32 WMMA, LDS double-buffered via async copies)
//           + multi-sigma exp kernel + weighted reduction.
// One block per upper-triangular 128x128 tile; 8 waves, each owning a 64x32
// patch = 4x2 grid of 16x16 accumulators.
// ---------------------------------------------------------------------------
__global__ void __launch_bounds__(256)
mmd_tile(const float* __restrict__ input,
         const float* __restrict__ target,
         const float* __restrict__ z2,
         const float* __restrict__ wcat,
         float* __restrict__ partials)
{
    __shared__ __align__(16) float smA[2][128 * LSTR];
    __shared__ __align__(16) float smB[2][128 * LSTR];
    __shared__ float sred[256];

    const int b = blockIdx.x / TRI_TILES;
    int t = blockIdx.x % TRI_TILES;
    int tm = 0, rowlen = TILES;                 // decode triangular index (uniform)
    while (t >= rowlen) { t -= rowlen; ++tm; --rowlen; }
    const int tn = tm + t;
    const bool offdiag = (tm != tn);

    const int tid  = threadIdx.x;
    const int wave = tid >> 5;
    const int lane = tid & 31;
    const int half = lane >> 4;
    const int l16  = lane & 15;

    // ---- staging addressing: thread -> (row = tid>>2 + pass*64, 16B col tid&3)
    const int sr = tid >> 2;
    const int sc = (tid & 3) * 4;
    const float* gA[2]; const float* gB[2]; int lofs[2];
#pragma unroll
    for (int pass = 0; pass < 2; ++pass) {
        const int r = sr + pass * 64;
        gA[pass] = zrow(input, target, b, tm * 128 + r) + sc;
        gB[pass] = zrow(input, target, b, tn * 128 + r) + sc;
        lofs[pass] = r * LSTR + sc;
    }

    // ---- WMMA fragment LDS offsets (A 16x4 f32 layout: half selects K pair)
    const int aoff = ((wave & 1) * 64 + l16) * LSTR + 2 * half;
    const int boff = ((wave >> 1) * 32 + l16) * LSTR + 2 * half;

    v8f acc[4][2];
#pragma unroll
    for (int i = 0; i < 4; ++i)
#pragma unroll
        for (int j = 0; j < 2; ++j) acc[i][j] = {};

    // prologue: stage chunk 0 into buffer 0
#pragma unroll
    for (int pass = 0; pass < 2; ++pass) {
        cp_b128(gA[pass], &smA[0][lofs[pass]]);
        if (offdiag) cp_b128(gB[pass], &smB[0][lofs[pass]]);
    }
    cp_commit_barrier();

    for (int kc = 0; kc < NC; ++kc) {
        const int cur = kc & 1;
        if (kc + 1 < NC) {                       // stage next chunk (other buffer)
            const int nxt  = cur ^ 1;
            const int gofs = (kc + 1) * KT;
#pragma unroll
            for (int pass = 0; pass < 2; ++pass) {
                cp_b128(gA[pass] + gofs, &smA[nxt][lofs[pass]]);
                if (offdiag) cp_b128(gB[pass] + gofs, &smB[nxt][lofs[pass]]);
            }
        }
        const float* As = &smA[cur][0];
        const float* Bs = offdiag ? &smB[cur][0] : &smA[cur][0];
#pragma unroll
        for (int kk = 0; kk < KT; kk += 4) {
            v2f af[4], bf[2];
#pragma unroll
            for (int i = 0; i < 4; ++i)
                af[i] = *(const v2f*)(As + aoff + i * (16 * LSTR) + kk);
#pragma unroll
            for (int j = 0; j < 2; ++j)
                bf[j] = *(const v2f*)(Bs + boff + j * (16 * LSTR) + kk);
#pragma unroll
            for (int i = 0; i < 4; ++i)
#pragma unroll
                for (int j = 0; j < 2; ++j)
                    acc[i][j] = __builtin_amdgcn_wmma_f32_16x16x4_f32(
                        false, af[i], false, bf[j], (short)0, acc[i][j],
                        false, false);
        }
        if (kc + 1 < NC) cp_commit_barrier();
    }

    // ---- epilogue: e = z_p.z_q + z2_p + z2_q ; sum_s exp(e/s) = t+t^2+t^4+t^8
    // C layout: VGPR r, lane -> (m = r + 8*half, n = l16).
    const int pwave = tm * 128 + (wave & 1) * 64;
    const int qwave = tn * 128 + (wave >> 1) * 32;
    const float* z2b = z2   + b * PP;
    const float* wb  = wcat + b * PP;
    float local = 0.f;
#pragma unroll
    for (int j = 0; j < 2; ++j) {
        const int q     = qwave + j * 16 + l16;
        const float z2q = z2b[q];
        const float wq  = wb[q];
#pragma unroll
        for (int i = 0; i < 4; ++i) {
            const int pbase = pwave + i * 16 + 8 * half;
#pragma unroll
            for (int r = 0; r < 8; ++r) {
                const int p    = pbase + r;
                const float e  = acc[i][j][r] + z2b[p] + z2q;
                const float t1 = __expf(e * 0.125f);
                const float t2 = t1 * t1;
                const float t4 = t2 * t2;
                const float t8 = t4 * t4;
                local += wb[p] * wq * (t1 + t2 + t4 + t8);
            }
        }
    }
    if (offdiag) local *= 2.f;   // tile stands for (tm,tn) and (tn,tm)

    // deterministic block reduction -> one partial per block
    sred[tid] = local;
    __syncthreads();
#pragma unroll
    for (int s = 128; s > 0; s >>= 1) {
        if (tid < s) sred[tid] += sred[tid + s];
        __syncthreads();
    }
    if (tid == 0) partials[blockIdx.x] = sred[0];
}

// ---------------------------------------------------------------------------
// Kernel 2: per-batch reduce of 136 partials, sqrt(kernel_sum + 1e-4), sum.
// ---------------------------------------------------------------------------
__global__ void mmd_final(const float* __restrict__ partials,
                          float* __restrict__ out)
{
    __shared__ float sred[256];
    const int tid = threadIdx.x;
    float total = 0.f;
    for (int b = 0; b < NB; ++b) {
        float v = 0.f;
        for (int t = tid; t < TRI_TILES; t += 256)
            v += partials[b * TRI_TILES + t];
        sred[tid] = v;
        __syncthreads();
#pragma unroll
        for (int s = 128; s > 0; s >>= 1) {
            if (tid < s) sred[tid] += sred[tid + s];
            __syncthreads();
        }
        if (tid == 0) total += sqrtf(sred[0] + 1e-4f);
        __syncthreads();
    }
    if (tid == 0) out[0] = total;
}

// ---------------------------------------------------------------------------
extern "C" void kernel_launch(void* const* d_in, const int* in_sizes, int n_in,
                              void* d_out, int out_size, void* d_ws, size_t ws_size,
                              hipStream_t stream)
{
    const float* input  = (const float*)d_in[0];  // (B, M, D)
    const float* target = (const float*)d_in[1];  // (B, N, D)
    const float* w_in   = (const float*)d_in[2];  // (B, M)
    const float* w_tar  = (const float*)d_in[3];  // (B, N)

    float* z2       = (float*)d_ws;          // NB*PP floats
    float* wcat     = z2 + NB * PP;          // NB*PP floats
    float* partials = wcat + NB * PP;        // NB*TRI_TILES floats

    mmd_prep<<<NB * PP / 8, 256, 0, stream>>>(input, target, w_in, w_tar, z2, wcat);
    mmd_tile<<<NB * TRI_TILES, 256, 0, stream>>>(input, target, z2, wcat, partials);
    mmd_final<<<1, 256, 0, stream>>>(partials, (float*)d_out);
}